// Attn_17059610099812
// MI455X (gfx1250) — compile-verified
//
#include <hip/hip_runtime.h>
#include <hip/hip_bf16.h>
#include <math.h>

// MI455X (gfx1250) attention-energy + softmax.
// Bandwidth-bound: 256MB of E streamed once (~11us floor at 23.3 TB/s).
// Dot products done with V_WMMA_F32_16X16X4_F32 (fp32 in/out, no cvt VALU).

#define SEQ_LEN 131072
#define HIDDEN  512
#define TILE_S  128          // rows per block (8 waves x 16 rows)
#define KC      64           // K-slab staged in LDS per iteration
#define LDSTR   (KC + 4)     // 68-dword row stride -> ds_load_b64 conflict-free

typedef __attribute__((ext_vector_type(2))) float v2f;
typedef __attribute__((ext_vector_type(8))) float v8f;

__global__ __launch_bounds__(256) void energies_kernel(
    const float* __restrict__ E,   // [SEQ_LEN, HIDDEN] row-major
    const float* __restrict__ h,   // [HIDDEN]
    float* __restrict__ eout,      // [SEQ_LEN] energies (aliases d_out)
    float* __restrict__ partials)  // [gridDim.x * 2] (max, sumexp) pairs
{
    __shared__ float tile[TILE_S * LDSTR];  // 34.0 KB
    __shared__ float hs[HIDDEN];            //  2.0 KB
    __shared__ float red[TILE_S];

    const int t    = threadIdx.x;
    const int w    = t >> 5;        // wave id 0..7
    const int lane = t & 31;
    const int half = lane >> 4;     // 0: K={0,1}, 1: K={2,3} (A/B VGPR layout)
    const int m    = lane & 15;     // row within the wave's 16-row group
    const int s0   = blockIdx.x * TILE_S;
    const int rowA = (w << 4) + m;  // tile row this lane reads for the A fragment

    // Stage hidden vector once (512 floats).
    if (t < HIDDEN / 4) {
        *(float4*)&hs[t << 2] = *(const float4*)&h[t << 2];
    }

    v8f acc = {0.f, 0.f, 0.f, 0.f, 0.f, 0.f, 0.f, 0.f};

    for (int ko = 0; ko < HIDDEN; ko += KC) {
        __syncthreads();  // previous compute done reading tile (also fences hs on iter 0)

        // Cooperative load: 128 rows x 64 floats = 2048 float4s, 8 per thread.
        #pragma unroll
        for (int j = 0; j < 8; ++j) {
            int i   = t + (j << 8);     // 0..2047
            int r   = i >> 4;           // row 0..127
            int seg = i & 15;           // 16B segment 0..15
            float4 v = *(const float4*)&E[(size_t)(s0 + r) * HIDDEN + ko + (seg << 2)];
            *(float4*)&tile[r * LDSTR + (seg << 2)] = v;
        }

        // Prefetch next K-slab (global_prefetch_b8): 128 rows x 2 cachelines.
        if (ko + KC < HIDDEN) {
            const float* pf =
                &E[(size_t)(s0 + (t & 127)) * HIDDEN + ko + KC + ((t >> 7) << 5)];
            __builtin_prefetch(pf, 0, 0);
        }

        __syncthreads();

        // 16 WMMAs per slab: D(16x16) += A(16x4, rows of E) x B(4x16, h replicated).
        #pragma unroll
        for (int c = 0; c < KC / 4; ++c) {
            int kk = c << 2;
            v2f a = *(const v2f*)&tile[rowA * LDSTR + kk + (half << 1)];
            v2f b = *(const v2f*)&hs[ko + kk + (half << 1)];
            acc = __builtin_amdgcn_wmma_f32_16x16x4_f32(
                false, a, false, b, (short)0, acc, false, false);
        }
    }

    // C layout: VGPR j -> M=j (lanes 0-15), M=j+8 (lanes 16-31); all N identical.
    if (m == 0) {
        #pragma unroll
        for (int j = 0; j < 8; ++j)
            red[(w << 4) + (half << 3) + j] = acc[j];
    }
    __syncthreads();

    if (t < TILE_S) eout[s0 + t] = red[t];

    // Wave 0: block-local online-softmax partial (max, sum of exp).
    if (w == 0) {
        float v0 = red[lane], v1 = red[lane + 32];
        float v2 = red[lane + 64], v3 = red[lane + 96];
        float mx = fmaxf(fmaxf(v0, v1), fmaxf(v2, v3));
        #pragma unroll
        for (int off = 16; off > 0; off >>= 1)
            mx = fmaxf(mx, __shfl_xor(mx, off, 32));
        float s = __expf(v0 - mx) + __expf(v1 - mx) +
                  __expf(v2 - mx) + __expf(v3 - mx);
        #pragma unroll
        for (int off = 16; off > 0; off >>= 1)
            s += __shfl_xor(s, off, 32);
        if (lane == 0) {
            partials[2 * blockIdx.x]     = mx;
            partials[2 * blockIdx.x + 1] = s;
        }
    }
}

__global__ __launch_bounds__(256) void reduce_kernel(
    const float* __restrict__ partials, float* __restrict__ msum, int n)
{
    __shared__ float sm[256], ss[256];
    const int t = threadIdx.x;
    float m = -__builtin_inff(), s = 0.f;
    for (int i = t; i < n; i += 256) {
        float mi = partials[2 * i], si = partials[2 * i + 1];
        float M = fmaxf(m, mi);
        s = s * __expf(m - M) + si * __expf(mi - M);
        m = M;
    }
    sm[t] = m; ss[t] = s;
    __syncthreads();
    for (int off = 128; off > 0; off >>= 1) {
        if (t < off) {
            float m2 = sm[t + off], s2 = ss[t + off];
            float M = fmaxf(sm[t], m2);
            ss[t] = ss[t] * __expf(sm[t] - M) + s2 * __expf(m2 - M);
            sm[t] = M;
        }
        __syncthreads();
    }
    if (t == 0) { msum[0] = sm[0]; msum[1] = 1.0f / ss[0]; }
}

__global__ __launch_bounds__(256) void normalize_kernel(
    float* __restrict__ out, const float* __restrict__ msum)
{
    const float M = msum[0], invS = msum[1];
    int i = (blockIdx.x * 256 + threadIdx.x) << 2;
    float4 v = *(float4*)&out[i];
    v.x = __expf(v.x - M) * invS;
    v.y = __expf(v.y - M) * invS;
    v.z = __expf(v.z - M) * invS;
    v.w = __expf(v.w - M) * invS;
    *(float4*)&out[i] = v;
}

extern "C" void kernel_launch(void* const* d_in, const int* in_sizes, int n_in,
                              void* d_out, int out_size, void* d_ws, size_t ws_size,
                              hipStream_t stream) {
    const float* hidden = (const float*)d_in[0];   // [512]
    const float* enc    = (const float*)d_in[1];   // [131072, 512]
    float* out  = (float*)d_out;                   // [131072]
    float* wsf  = (float*)d_ws;
    float* msum     = wsf;       // 2 floats: M, 1/S
    float* partials = wsf + 2;   // 1024 (m,s) pairs = 8 KB

    const int nblk = SEQ_LEN / TILE_S;  // 1024
    energies_kernel<<<nblk, 256, 0, stream>>>(enc, hidden, out, partials);
    reduce_kernel<<<1, 256, 0, stream>>>(partials, msum, nblk);
    normalize_kernel<<<SEQ_LEN / (256 * 4), 256, 0, stream>>>(out, msum);
}